// PlanarFlow_54623394070880
// MI455X (gfx1250) — compile-verified
//
#include <hip/hip_runtime.h>

typedef __attribute__((ext_vector_type(2))) float v2f;
typedef __attribute__((ext_vector_type(8))) float v8f;

#define DIMS 64

__global__ __launch_bounds__(256) void planar_flow_wmma_kernel(
    const float* __restrict__ z, const float* __restrict__ W0,
    const float* __restrict__ b0, const float* __restrict__ W1,
    float* __restrict__ xout, float* __restrict__ ldout, int rows)
{
    const int lane = threadIdx.x & 31;
    const int wave = threadIdx.x >> 5;
    const int hh   = lane >> 4;      // which 16-lane half
    const int n    = lane & 15;      // column within a 16-wide tile
    const int rowbase = (blockIdx.x * 8 + wave) * 16;
    if (rowbase >= rows) return;     // wave-uniform: EXEC stays all-ones

    // Per-lane slices of W0 / W1 (column = 16*c + n). Small, hot in L2.
    float w0c[4], w1c[4];
#pragma unroll
    for (int c = 0; c < 4; ++c) {
        w0c[c] = W0[16 * c + n];
        w1c[c] = W1[16 * c + n];
    }
    const float bias = b0[0];

    // Load z tile directly in the 16x16 f32 C/D WMMA layout:
    // zc[c][v] = z[rowbase + v + 8*hh][16*c + n]
    float zc[4][8];
#pragma unroll
    for (int c = 0; c < 4; ++c) {
#pragma unroll
        for (int v = 0; v < 8; ++v) {
            const long long r = (long long)(rowbase + v + 8 * hh);
            zc[c][v] = __builtin_nontemporal_load(&z[r * DIMS + 16 * c + n]);
        }
    }

    // a[M] = z_row . W0 ; reduce partials across the 16 lanes of each half.
    // Low-half lanes end up with rows 0..7, high-half lanes rows 8..15.
    float t[8];
#pragma unroll
    for (int v = 0; v < 8; ++v) {
        float p = zc[0][v] * w0c[0] + zc[1][v] * w0c[1]
                + zc[2][v] * w0c[2] + zc[3][v] * w0c[3];
        p += __shfl_xor(p, 1, 32);
        p += __shfl_xor(p, 2, 32);
        p += __shfl_xor(p, 4, 32);
        p += __shfl_xor(p, 8, 32);
        t[v] = tanhf(p + bias);
    }

    // s = W0 . W1 (each half's 16 lanes cover all 64 columns)
    float s = w0c[0] * w1c[0] + w0c[1] * w1c[1]
            + w0c[2] * w1c[2] + w0c[3] * w1c[3];
    s += __shfl_xor(s, 1, 32);
    s += __shfl_xor(s, 2, 32);
    s += __shfl_xor(s, 4, 32);
    s += __shfl_xor(s, 8, 32);

    // Build A operand: lane l must hold t[row = l & 15].
    float tOther[8];
#pragma unroll
    for (int v = 0; v < 8; ++v) tOther[v] = __shfl_xor(t[v], 16, 32);
    float tlo[8], thi[8];                    // tlo[v] = t(row v), thi[v] = t(row 8+v)
#pragma unroll
    for (int v = 0; v < 8; ++v) {
        tlo[v] = hh ? tOther[v] : t[v];
        thi[v] = hh ? t[v]      : tOther[v];
    }
    float tA = tlo[0];
#pragma unroll
    for (int v = 1; v < 8; ++v) tA = (n == v) ? tlo[v] : tA;
#pragma unroll
    for (int v = 0; v < 8; ++v) tA = (n == 8 + v) ? thi[v] : tA;

    // Replicate t across all K slots of A; W1/4 across all K slots of B, so
    // D = sum_K t[M] * W1[N]/4 = t[M]*W1[N], independent of K<->slot mapping.
    v2f A; A.x = tA; A.y = tA;

#pragma unroll
    for (int c = 0; c < 4; ++c) {
        const float bq = 0.25f * w1c[c];
        v2f B; B.x = bq; B.y = bq;
        v8f C;
#pragma unroll
        for (int v = 0; v < 8; ++v) C[v] = zc[c][v];
        // x_tile = t (outer) W1_chunk + z_tile  (residual fused via C operand)
        v8f D = __builtin_amdgcn_wmma_f32_16x16x4_f32(
            /*neg_a=*/false, A, /*neg_b=*/false, B,
            /*c_mod=*/(short)0, C, /*reuse_a=*/false, /*reuse_b=*/false);
#pragma unroll
        for (int v = 0; v < 8; ++v) {
            const long long r = (long long)(rowbase + v + 8 * hh);
            __builtin_nontemporal_store(D[v], &xout[r * DIMS + 16 * c + n]);
        }
    }

    // log_det: det = 1 + (1 - t^2) * (W0.W1); one lane per half stores 8 rows.
    if (n == 0) {
#pragma unroll
        for (int v = 0; v < 8; ++v) {
            const int r = rowbase + v + 8 * hh;
            const float det = 1.0f + (1.0f - t[v] * t[v]) * s;
            __builtin_nontemporal_store(logf(fabsf(det)), &ldout[r]);
        }
    }
}

extern "C" void kernel_launch(void* const* d_in, const int* in_sizes, int n_in,
                              void* d_out, int out_size, void* d_ws, size_t ws_size,
                              hipStream_t stream) {
    const float* z  = (const float*)d_in[0];
    const float* W0 = (const float*)d_in[1];
    const float* b0 = (const float*)d_in[2];
    const float* W1 = (const float*)d_in[3];

    const int rows = in_sizes[0] / DIMS;   // N = 1048576 (multiple of 128)
    float* xout  = (float*)d_out;
    float* ldout = xout + (size_t)rows * DIMS;

    const int waves  = (rows + 15) / 16;   // 16 rows per wave
    const int blocks = (waves + 7) / 8;    // 8 waves (256 threads) per block

    planar_flow_wmma_kernel<<<blocks, 256, 0, stream>>>(z, W0, b0, W1, xout, ldout, rows);
}